// UG_GraphAttentionLayerBS_24343874634221
// MI455X (gfx1250) — compile-verified
//
#include <hip/hip_runtime.h>
#include <hip/hip_bf16.h>

#define Bb   8
#define Nn   2048
#define FIN  256
#define FOUT 128
#define BN   (Bb * Nn)          // 16384
#define ALPHA 0.2f
#define NEG_INF (-9.0e15f)
#define NEG_HUGE (-3.4e38f)
#define PART 16

typedef __attribute__((ext_vector_type(2))) float v2f;
typedef __attribute__((ext_vector_type(8))) float v8f;

// e[b,i,j] = adj>0 ? unc * leaky_relu(f1_i + f2_j) : -9e15
__device__ __forceinline__ float edge_e(int ad, float u, float f1i, float f2j) {
  float t = f1i + f2j;
  t = t > 0.f ? t : ALPHA * t;
  return (ad > 0) ? u * t : NEG_INF;
}

// one-exp online softmax update for a running (max m, sum s)
__device__ __forceinline__ void online_upd(float e, float& m, float& s) {
  const float nm = fmaxf(m, e);
  const float d  = __expf(fminf(m, e) - nm);   // exp(smaller - larger)
  s = (e > m) ? (s * d + 1.f) : (s + d);
  m = nm;
}

// ---------------------------------------------------------------------------
// Kernel 1: H = input @ W   (16384x256) @ (256x128), f32 WMMA 16x16x4
// block = 128 threads (4 waves); each wave -> 16 rows x 128 cols; tile 64x128
// ---------------------------------------------------------------------------
__global__ __launch_bounds__(128) void gemm_h_kernel(
    const float* __restrict__ X, const float* __restrict__ W,
    float* __restrict__ H) {
  const int tid  = threadIdx.x;
  const int wave = tid >> 5;
  const int lane = tid & 31;
  const int nl   = lane & 15;          // m for A, n for B/D
  const int kp   = (lane >> 4) << 1;   // 0 or 2 (K sub-pair)
  const int row0 = blockIdx.x * 64 + wave * 16;

  const float* __restrict__ Arow = X + (size_t)(row0 + nl) * FIN;

  v8f acc[8] = {};
  for (int k = 0; k < FIN; k += 4) {
    v2f a;
    a.x = Arow[k + kp];
    a.y = Arow[k + kp + 1];
#pragma unroll
    for (int t = 0; t < 8; ++t) {
      v2f bv;
      const int n = t * 16 + nl;
      bv.x = W[(size_t)(k + kp) * FOUT + n];
      bv.y = W[(size_t)(k + kp + 1) * FOUT + n];
      acc[t] = __builtin_amdgcn_wmma_f32_16x16x4_f32(
          false, a, false, bv, (short)0, acc[t], false, false);
    }
  }
#pragma unroll
  for (int r = 0; r < 8; ++r) {
    const int mr = (lane >> 4) * 8 + r;
#pragma unroll
    for (int t = 0; t < 8; ++t)
      H[(size_t)(row0 + mr) * FOUT + t * 16 + nl] = acc[t][r];
  }
}

// ---------------------------------------------------------------------------
// Kernel 2: f1[bn] = h[bn,:]·a1, f2[bn] = h[bn,:]·a2. One wave per row,
// float4 (b128) coalesced reads.
// ---------------------------------------------------------------------------
__global__ __launch_bounds__(256) void f12_kernel(
    const float* __restrict__ H, const float* __restrict__ a,
    float* __restrict__ f1, float* __restrict__ f2) {
  const int row  = blockIdx.x * 8 + (threadIdx.x >> 5);
  const int lane = threadIdx.x & 31;
  const float* __restrict__ hr = H + (size_t)row * FOUT;
  const float4 hv = *(const float4*)(hr + lane * 4);
  const float4 av = *(const float4*)(a + lane * 4);
  const float4 bv = *(const float4*)(a + FOUT + lane * 4);
  float s1 = hv.x * av.x + hv.y * av.y + hv.z * av.z + hv.w * av.w;
  float s2 = hv.x * bv.x + hv.y * bv.y + hv.z * bv.z + hv.w * bv.w;
#pragma unroll
  for (int off = 16; off > 0; off >>= 1) {
    s1 += __shfl_xor(s1, off, 32);
    s2 += __shfl_xor(s2, off, 32);
  }
  if (lane == 0) { f1[row] = s1; f2[row] = s2; }
}

// ---------------------------------------------------------------------------
// Kernel 3: per-(b,j) partial column softmax stats over an i-chunk.
// 4 columns per thread, int4/float4 (b128) streaming loads, 1 exp/element.
// grid = (N/1024, B, PART)
// ---------------------------------------------------------------------------
__global__ __launch_bounds__(256) void colpart_kernel(
    const int* __restrict__ adj, const float* __restrict__ unc,
    const float* __restrict__ f1, const float* __restrict__ f2,
    float* __restrict__ pM, float* __restrict__ pS) {
  const int b    = blockIdx.y;
  const int part = blockIdx.z;
  const int j4   = (blockIdx.x * 256 + threadIdx.x) * 4;
  const int i0   = part * (Nn / PART);
  const size_t base = (size_t)b * Nn;
  const float4 f2v = *(const float4*)(f2 + base + j4);

  float m0 = NEG_HUGE, m1 = NEG_HUGE, m2 = NEG_HUGE, m3 = NEG_HUGE;
  float s0 = 0.f, s1 = 0.f, s2 = 0.f, s3 = 0.f;
  for (int i = i0; i < i0 + Nn / PART; ++i) {
    const size_t g = (base + i) * Nn + j4;
    const int4   ad = *(const int4*)(adj + g);
    const float4 uv = *(const float4*)(unc + g);
    const float f1i = f1[base + i];
    online_upd(edge_e(ad.x, uv.x, f1i, f2v.x), m0, s0);
    online_upd(edge_e(ad.y, uv.y, f1i, f2v.y), m1, s1);
    online_upd(edge_e(ad.z, uv.z, f1i, f2v.z), m2, s2);
    online_upd(edge_e(ad.w, uv.w, f1i, f2v.w), m3, s3);
  }
  const size_t o = (size_t)part * BN + base + j4;
  pM[o + 0] = m0; pS[o + 0] = s0;
  pM[o + 1] = m1; pS[o + 1] = s1;
  pM[o + 2] = m2; pS[o + 2] = s2;
  pM[o + 3] = m3; pS[o + 3] = s3;
}

// ---------------------------------------------------------------------------
// Kernel 4: combine partial stats -> column max M and 1/S
// ---------------------------------------------------------------------------
__global__ __launch_bounds__(256) void colcomb_kernel(
    const float* __restrict__ pM, const float* __restrict__ pS,
    float* __restrict__ colM, float* __restrict__ colInvS) {
  const int bn = blockIdx.x * 256 + threadIdx.x;
  float m = NEG_HUGE;
#pragma unroll
  for (int p = 0; p < PART; ++p) m = fmaxf(m, pM[(size_t)p * BN + bn]);
  float s = 0.f;
#pragma unroll
  for (int p = 0; p < PART; ++p)
    s += pS[(size_t)p * BN + bn] * __expf(pM[(size_t)p * BN + bn] - m);
  colM[bn]    = m;
  colInvS[bn] = 1.0f / s;
}

// ---------------------------------------------------------------------------
// Kernel 5: out = elu( P @ (H * invS) ), P[b,i,j] = exp(e[b,i,j]-M[b,j])
// block = 128 threads (4 waves); tile 64 (i) x 128 (f); K(j)-chunk = 32.
// b128 staging into LDS; f32 WMMA 16x16x4 inner product.
//   sP stride 36: 36*d mod 64 distinct for d=0..15 -> conflict-free A frags,
//                 rows stay 8B-aligned for b64 A reads, 16B-aligned stores.
//   sH stride 136: 2*136 = 16 mod 64 -> half-wave row groups use disjoint
//                  bank ranges for B frags, 16B-aligned stores.
// ---------------------------------------------------------------------------
__global__ __launch_bounds__(128) void attn_kernel(
    const int* __restrict__ adj, const float* __restrict__ unc,
    const float* __restrict__ f1, const float* __restrict__ f2,
    const float* __restrict__ colM, const float* __restrict__ colInvS,
    const float* __restrict__ H, float* __restrict__ out) {
  __shared__ __align__(16) float sP[64][36];    // P tile (i x j)
  __shared__ __align__(16) float sH[32][136];   // scaled-H tile (j x f)

  const int tid  = threadIdx.x;
  const int wave = tid >> 5;
  const int lane = tid & 31;
  const int nl   = lane & 15;
  const int kp   = (lane >> 4) << 1;
  const int b    = blockIdx.y;
  const int i0   = blockIdx.x * 64;
  const size_t bb = (size_t)b * Nn;

  v8f acc[8] = {};
  for (int jc = 0; jc < Nn; jc += 32) {
    // stage P tile 64x32: 4 consecutive j per thread (b128 loads)
#pragma unroll
    for (int rep = 0; rep < 4; ++rep) {
      const int idx = tid + rep * 128;        // 0..511
      const int ii  = idx >> 3;               // 0..63
      const int jj4 = (idx & 7) << 2;         // 0,4,...,28
      const size_t g = (bb + i0 + ii) * Nn + jc + jj4;
      const int4   ad  = *(const int4*)(adj + g);
      const float4 uv  = *(const float4*)(unc + g);
      const float4 f2v = *(const float4*)(f2 + bb + jc + jj4);
      const float4 mv  = *(const float4*)(colM + bb + jc + jj4);
      const float  f1i = f1[bb + i0 + ii];
      float4 p;
      p.x = __expf(edge_e(ad.x, uv.x, f1i, f2v.x) - mv.x);
      p.y = __expf(edge_e(ad.y, uv.y, f1i, f2v.y) - mv.y);
      p.z = __expf(edge_e(ad.z, uv.z, f1i, f2v.z) - mv.z);
      p.w = __expf(edge_e(ad.w, uv.w, f1i, f2v.w) - mv.w);
      *(float4*)&sP[ii][jj4] = p;
    }
    // stage scaled-H tile 32x128 (b128 loads); fold in 1/S
#pragma unroll
    for (int rep = 0; rep < 8; ++rep) {
      const int idx = tid + rep * 128;        // 0..1023
      const int jj  = idx >> 5;               // 0..31
      const int ff4 = (idx & 31) << 2;        // 0..124
      float4 hv = *(const float4*)(H + (bb + jc + jj) * FOUT + ff4);
      const float is = colInvS[bb + jc + jj];
      hv.x *= is; hv.y *= is; hv.z *= is; hv.w *= is;
      *(float4*)&sH[jj][ff4] = hv;
    }
    __syncthreads();

    const int mrow = wave * 16 + nl;
#pragma unroll
    for (int kk = 0; kk < 32; kk += 4) {
      v2f a;
      a.x = sP[mrow][kk + kp];
      a.y = sP[mrow][kk + kp + 1];
#pragma unroll
      for (int t = 0; t < 8; ++t) {
        v2f bv;
        bv.x = sH[kk + kp][t * 16 + nl];
        bv.y = sH[kk + kp + 1][t * 16 + nl];
        acc[t] = __builtin_amdgcn_wmma_f32_16x16x4_f32(
            false, a, false, bv, (short)0, acc[t], false, false);
      }
    }
    __syncthreads();
  }

  // ELU epilogue + store
#pragma unroll
  for (int r = 0; r < 8; ++r) {
    const int mr = (lane >> 4) * 8 + r;
    const int gi = i0 + wave * 16 + mr;
#pragma unroll
    for (int t = 0; t < 8; ++t) {
      const float v = acc[t][r];
      const float o = v > 0.f ? v : (__expf(v) - 1.0f);
      out[(bb + gi) * FOUT + t * 16 + nl] = o;
    }
  }
}

// ---------------------------------------------------------------------------
extern "C" void kernel_launch(void* const* d_in, const int* in_sizes, int n_in,
                              void* d_out, int out_size, void* d_ws, size_t ws_size,
                              hipStream_t stream) {
  (void)in_sizes; (void)n_in; (void)out_size; (void)ws_size;
  const float* input = (const float*)d_in[0];   // (B,N,FIN)
  const int*   adj   = (const int*)d_in[1];     // (B,N,N)
  const float* unc   = (const float*)d_in[2];   // (B,N,N)
  const float* W     = (const float*)d_in[3];   // (FIN,FOUT)
  const float* a     = (const float*)d_in[4];   // (2*FOUT,1)
  float* out = (float*)d_out;                   // (B,N,FOUT)

  float* ws      = (float*)d_ws;
  float* f1      = ws;                          // BN
  float* f2      = f1 + BN;                     // BN
  float* colM    = f2 + BN;                     // BN
  float* colInvS = colM + BN;                   // BN
  float* H       = colInvS + BN;                // BN*FOUT
  float* pM      = H + (size_t)BN * FOUT;       // PART*BN
  float* pS      = pM + (size_t)PART * BN;      // PART*BN

  gemm_h_kernel<<<BN / 64, 128, 0, stream>>>(input, W, H);
  f12_kernel<<<BN / 8, 256, 0, stream>>>(H, a, f1, f2);
  colpart_kernel<<<dim3(Nn / 1024, Bb, PART), 256, 0, stream>>>(adj, unc, f1, f2, pM, pS);
  colcomb_kernel<<<BN / 256, 256, 0, stream>>>(pM, pS, colM, colInvS);
  attn_kernel<<<dim3(Nn / 64, Bb), 128, 0, stream>>>(adj, unc, f1, f2, colM, colInvS, H, out);
}